// AUAvULoss_61040075210978
// MI455X (gfx1250) — compile-verified
//
#include <hip/hip_runtime.h>
#include <math.h>

#define EPSF    1e-10f
#define N_TH    21
#define BLOCKS1 4096
#define BLOCKS2 1024
#define PSTRIDE 88   // 4*21 counters + 2 totals, padded

typedef float v2f __attribute__((ext_vector_type(2)));
typedef float v8f __attribute__((ext_vector_type(8)));

// fast reciprocal: v_rcp_f32 (~1 ulp), avoids IEEE div_scale/div_fmas expansion
__device__ __forceinline__ float fast_rcp(float x) {
  return __builtin_amdgcn_rcpf(x);
}

// branchless tanh for u >= 0 (entropy): tanh(u) = (1-e)/(1+e), e = exp(-2u)
__device__ __forceinline__ float fast_tanh_pos(float u) {
  float e = __expf(-2.0f * u);
  return (1.0f - e) * fast_rcp(1.0f + e);
}

// ---------------- Pass 1: per-row softmax stats (wave32 per row) ----------------
__global__ void __launch_bounds__(256) avu_rows_kernel(
    const float* __restrict__ logits, const int* __restrict__ labels,
    float2* __restrict__ rows, float* __restrict__ bmin, float* __restrict__ bmax,
    int N, int C4)
{
  const int lane = threadIdx.x & 31;
  const int wib  = threadIdx.x >> 5;
  // scalarize the wave id so the row loop is SALU-controlled (EXEC stays full)
  const int wave = __builtin_amdgcn_readfirstlane((blockIdx.x * blockDim.x + threadIdx.x) >> 5);
  const int nwav = (gridDim.x * blockDim.x) >> 5;
  const int C    = C4 * 4;
  const bool act = lane < C4;

  float lmin = INFINITY, lmax = -INFINITY;

  for (int r = wave; r < N; r += nwav) {
    const float4* rp = reinterpret_cast<const float4*>(logits + (size_t)r * C);
    float4 x = make_float4(-INFINITY, -INFINITY, -INFINITY, -INFINITY);
    if (act) x = rp[lane];                       // one coalesced 400B row read
    if (r + nwav < N)
      __builtin_prefetch(logits + (size_t)(r + nwav) * C + lane * 4, 0, 1);

    // row max + argmax (first occurrence)
    float m = x.x; int mi = 0;
    if (x.y > m) { m = x.y; mi = 1; }
    if (x.z > m) { m = x.z; mi = 2; }
    if (x.w > m) { m = x.w; mi = 3; }
    mi += lane * 4;
#pragma unroll
    for (int off = 16; off > 0; off >>= 1) {
      float om = __shfl_xor(m, off, 32);
      int   oi = __shfl_xor(mi, off, 32);
      if (om > m || (om == m && oi < mi)) { m = om; mi = oi; }
    }

    // S = sum(exp(x-m)),  T = sum((x-m)*exp(x-m))
    float s = 0.f, tq = 0.f;
    if (act) {
      float d0 = x.x - m, d1 = x.y - m, d2 = x.z - m, d3 = x.w - m;
      float e0 = __expf(d0), e1 = __expf(d1), e2 = __expf(d2), e3 = __expf(d3);
      s  = (e0 + e1) + (e2 + e3);
      tq = (d0 * e0 + d1 * e1) + (d2 * e2 + d3 * e3);
    }
#pragma unroll
    for (int off = 16; off > 0; off >>= 1) {
      s  += __shfl_xor(s,  off, 32);
      tq += __shfl_xor(tq, off, 32);
    }

    float rs   = fast_rcp(s);              // one v_rcp, reused
    float conf = rs;                       // max prob = exp(0)/S
    float unc  = __logf(s) - tq * rs;      // entropy (eps term ~1e-8 abs, negligible)
    lmin = fminf(lmin, unc);
    lmax = fmaxf(lmax, unc);

    if (lane == 0)
      rows[r] = make_float2(unc, (labels[r] == mi) ? conf : -conf); // sign = accurate
  }

  __shared__ float smin[8], smax[8];
  if (lane == 0) { smin[wib] = lmin; smax[wib] = lmax; }
  __syncthreads();
  if (threadIdx.x == 0) {
    float mn = smin[0], mx = smax[0];
#pragma unroll
    for (int i = 1; i < 8; ++i) { mn = fminf(mn, smin[i]); mx = fmaxf(mx, smax[i]); }
    bmin[blockIdx.x] = mn; bmax[blockIdx.x] = mx;
  }
}

// ---------------- Pass 1b: reduce per-block min/max ----------------
__global__ void __launch_bounds__(256) avu_minmax_kernel(
    const float* __restrict__ bmin, const float* __restrict__ bmax,
    float* __restrict__ minmax, int nb)
{
  float mn = INFINITY, mx = -INFINITY;
  for (int i = threadIdx.x; i < nb; i += blockDim.x) {
    mn = fminf(mn, bmin[i]); mx = fmaxf(mx, bmax[i]);
  }
#pragma unroll
  for (int off = 16; off > 0; off >>= 1) {
    mn = fminf(mn, __shfl_xor(mn, off, 32));
    mx = fmaxf(mx, __shfl_xor(mx, off, 32));
  }
  __shared__ float smin[8], smax[8];
  int wib = threadIdx.x >> 5;
  if ((threadIdx.x & 31) == 0) { smin[wib] = mn; smax[wib] = mx; }
  __syncthreads();
  if (threadIdx.x == 0) {
    float a = smin[0], b = smax[0];
    for (int i = 1; i < 8; ++i) { a = fminf(a, smin[i]); b = fmaxf(b, smax[i]); }
    minmax[0] = a; minmax[1] = b;
  }
}

// ---------------- Pass 2: cert[21,N] @ W[N,4] via V_WMMA_F32_16X16X4_F32 ----------------
__global__ void __launch_bounds__(256) avu_hist_kernel(
    const float2* __restrict__ rows, const float* __restrict__ minmax,
    float* __restrict__ partials, int N)
{
  const int lane = threadIdx.x & 31;
  const int half = lane >> 4;        // K-group / M-group selector per ISA layout
  const int lm   = lane & 15;
  const int wib  = threadIdx.x >> 5;
  const int wave = __builtin_amdgcn_readfirstlane((blockIdx.x * blockDim.x + threadIdx.x) >> 5);
  const int nwav = (gridDim.x * blockDim.x) >> 5;

  const float umin  = minmax[0];
  const float range = minmax[1] - umin;
  // unc_th for this lane's two A-matrix rows (M = lm and M = lm+16)
  const float uth0 = umin + ((float)lm * 0.05f) * range;
  const float uth1 = umin + ((float)(lm + 16) * 0.05f) * range;

  v8f c0 = {0.f,0.f,0.f,0.f,0.f,0.f,0.f,0.f};   // thresholds 0..15
  v8f c1 = {0.f,0.f,0.f,0.f,0.f,0.f,0.f,0.f};   // thresholds 16..20 (rest ignored)
  float tot_au = 0.f, tot_iu = 0.f;

  const int nfull = N >> 2;                      // full 4-row chunks, no guards
  for (int ch = wave; ch < nfull; ch += nwav) {
    const int base = ch << 2;
    const float4* rp = reinterpret_cast<const float4*>(rows + base);
    float4 q0 = rp[0], q1 = rp[1];               // 4 rows = 2 x b128 (scalarized broadcast)
    float u0 = q0.x, u1 = q0.z, u2 = q1.x, u3 = q1.z;
    float cw0 = q0.y, cw1 = q0.w, cw2 = q1.y, cw3 = q1.w;

#define ROW_W(U, CW, WA, WI, WU, WX)                         \
    float WA, WI, WU, WX;                                    \
    {                                                        \
      float conf = fabsf(CW);                                \
      float tu   = fast_tanh_pos(U);                         \
      float omt  = 1.0f - tu;                                \
      float pa   = (CW > 0.0f) ? conf : 0.0f;                \
      float pi   = (CW > 0.0f) ? 0.0f : (1.0f - conf);       \
      WA = pa * omt;  WI = pi * omt;                         \
      WU = pa * tu;   WX = pi * tu;                          \
    }
    ROW_W(u0, cw0, wa0, wi0, wu0, wx0)
    ROW_W(u1, cw1, wa1, wi1, wu1, wx1)
    ROW_W(u2, cw2, wa2, wi2, wu2, wx2)
    ROW_W(u3, cw3, wa3, wi3, wu3, wx3)
#undef ROW_W
    tot_au += (wu0 + wu1) + (wu2 + wu3);
    tot_iu += (wx0 + wx1) + (wx2 + wx3);

    // this lane's two K rows: K = 2*half, 2*half+1
    float uA  = half ? u2  : u0,  uB  = half ? u3  : u1;
    float waA = half ? wa2 : wa0, waB = half ? wa3 : wa1;
    float wiA = half ? wi2 : wi0, wiB = half ? wi3 : wi1;
    float wuA = half ? wu2 : wu0, wuB = half ? wu3 : wu1;
    float wxA = half ? wx2 : wx0, wxB = half ? wx3 : wx1;

    // A (16x4 cert matrix): lane = M, VGPR = K within half-wave group
    v2f a0 = { uA <= uth0 ? 1.f : 0.f, uB <= uth0 ? 1.f : 0.f };
    v2f a1 = { uA <= uth1 ? 1.f : 0.f, uB <= uth1 ? 1.f : 0.f };
    // B (4x16 weights): lane = N (cols 0..3 = ac,ic,au,iu), VGPR = K
    float bA = (lm == 0) ? waA : (lm == 1) ? wiA : (lm == 2) ? wuA : (lm == 3) ? wxA : 0.f;
    float bB = (lm == 0) ? waB : (lm == 1) ? wiB : (lm == 2) ? wuB : (lm == 3) ? wxB : 0.f;
    v2f b = { bA, bB };

    c0 = __builtin_amdgcn_wmma_f32_16x16x4_f32(false, a0, false, b, (short)0, c0, false, false);
    c1 = __builtin_amdgcn_wmma_f32_16x16x4_f32(false, a1, false, b, (short)0, c1, false, false);
  }

  // deterministic block reduction: each wave owns one LDS row, then in-order sum over the 8 rows
  __shared__ float sacc[8][PSTRIDE];
  for (int i = lane; i < PSTRIDE; i += 32) sacc[wib][i] = 0.f;  // wave-ordered with writes below
  if (lm < 4) {
#pragma unroll
    for (int g = 0; g < 8; ++g) {
      int t0 = g + 8 * half;                       // D layout: M = g + 8*(lane>=16)
      sacc[wib][lm * N_TH + t0] = c0[g];
      int t1 = t0 + 16;
      if (t1 < N_TH) sacc[wib][lm * N_TH + t1] = c1[g];
    }
  }
  if (lane == 0) { sacc[wib][84] = tot_au; sacc[wib][85] = tot_iu; }
  __syncthreads();
  if (threadIdx.x < 86) {
    float v = 0.f;
#pragma unroll
    for (int wq = 0; wq < 8; ++wq) v += sacc[wq][threadIdx.x];
    partials[blockIdx.x * PSTRIDE + threadIdx.x] = v;
  }
}

// ---------------- Pass 3: reduce partials, fold tail rows, AvU curve, AUC, loss ----------------
__global__ void __launch_bounds__(128) avu_final_kernel(
    const float* __restrict__ partials, const float2* __restrict__ rows,
    const float* __restrict__ minmax, float* __restrict__ out,
    int nblocks, int tail_start, int N)
{
  __shared__ float res[86];
  int c = threadIdx.x;
  if (c < 86) {
    float v = 0.f;
    for (int b = 0; b < nblocks; ++b) v += partials[b * PSTRIDE + c];
    res[c] = v;
  }
  __syncthreads();
  if (c == 0) {
    // scalar fold of up to 3 tail rows (N % 4)
    float umin = minmax[0], range = minmax[1] - umin;
    for (int r = tail_start; r < N; ++r) {
      float2 rj  = rows[r];
      float conf = fabsf(rj.y);
      float tu   = fast_tanh_pos(rj.x);
      bool  acc  = rj.y > 0.f;
      float pa   = acc ? conf : 0.f;
      float pi   = acc ? 0.f  : (1.f - conf);
      float wa_ = pa * (1.f - tu), wi_ = pi * (1.f - tu);
      float wu_ = pa * tu,         wx_ = pi * tu;
      res[84] += wu_; res[85] += wx_;
      for (int t = 0; t < N_TH; ++t) {
        float uth = umin + ((float)t * 0.05f) * range;
        if (rj.x <= uth) {
          res[0 * N_TH + t] += wa_; res[1 * N_TH + t] += wi_;
          res[2 * N_TH + t] += wu_; res[3 * N_TH + t] += wx_;
        }
      }
    }

    float tau = res[84], tiu = res[85];
    float avu[N_TH];
    for (int t = 0; t < N_TH; ++t) {
      float n_ac = res[0 * N_TH + t];
      float n_ic = res[1 * N_TH + t];
      float n_au = tau - res[2 * N_TH + t];   // (1-cert) side = total - cert side
      float n_iu = tiu - res[3 * N_TH + t];
      avu[t] = (n_ac + n_iu) / (((n_ac + n_au) + (n_ic + n_iu)) + EPSF);
    }
    float auc = 0.f;
    for (int t = 0; t < N_TH - 1; ++t) {
      float dth = (float)(t + 1) * 0.05f - (float)t * 0.05f;
      auc += 0.5f * (avu[t + 1] + avu[t]) * dth;
    }
    out[0] = -logf(auc + EPSF);   // avu_loss (BETA = 1)
    out[1] = auc;                 // auc_avu
  }
}

extern "C" void kernel_launch(void* const* d_in, const int* in_sizes, int n_in,
                              void* d_out, int out_size, void* d_ws, size_t ws_size,
                              hipStream_t stream)
{
  (void)n_in; (void)out_size; (void)ws_size;
  const float* logits = (const float*)d_in[0];
  const int*   labels = (const int*)d_in[1];   // jnp int64 demotes to int32 under default jax
  const int N  = in_sizes[1];
  const int C  = in_sizes[0] / N;              // 100
  const int C4 = C / 4;                        // 25 float4 per row

  char* ws = (char*)d_ws;
  float2* rows  = (float2*)ws;           size_t off = (size_t)N * sizeof(float2);
  float* bmin   = (float*)(ws + off);    off += (size_t)BLOCKS1 * sizeof(float);
  float* bmax   = (float*)(ws + off);    off += (size_t)BLOCKS1 * sizeof(float);
  float* minmax = (float*)(ws + off);    off += 2 * sizeof(float);
  float* parts  = (float*)(ws + off);    off += (size_t)BLOCKS2 * PSTRIDE * sizeof(float);

  const int tail_start = (N >> 2) << 2;

  avu_rows_kernel  <<<BLOCKS1, 256, 0, stream>>>(logits, labels, rows, bmin, bmax, N, C4);
  avu_minmax_kernel<<<1,       256, 0, stream>>>(bmin, bmax, minmax, BLOCKS1);
  avu_hist_kernel  <<<BLOCKS2, 256, 0, stream>>>(rows, minmax, parts, N);
  avu_final_kernel <<<1,       128, 0, stream>>>(parts, rows, minmax, (float*)d_out,
                                                 BLOCKS2, tail_start, N);
}